// GIN_4layer_basic_71949292143005
// MI455X (gfx1250) — compile-verified
//
#include <hip/hip_runtime.h>

// GIN 4-layer forward for MI455X (gfx1250, wave32).
// Per layer:  agg = segment_sum(h[src], dst);  h' = act((h + agg) @ W^T + b)
//
// Roofline: scatter dominates (~0.8 GB of gather + f32-atomic traffic per
// layer); feature buffers (25.6 MB) and edge list (12.8 MB) fit the 192 MB L2,
// so atomics resolve in L2. GEMM (0.82 GFLOP/layer) runs on the matrix pipe
// via V_WMMA_F32_16X16X4_F32 — full f32 precision, matching the reference.

#define NN 100000       // nodes  (== 6250 * 16, exact M tiles)
#define NE 1600000      // edges
#define FEAT 64
#define CLASSES 10

typedef __attribute__((ext_vector_type(2))) float v2f;
typedef __attribute__((ext_vector_type(8))) float v8f;

// ---------------------------------------------------------------------------
// Edge scatter: one wave per edge; lane l handles features {2l, 2l+1}.
// Edge indices are wave-uniform -> readfirstlane promotes them to SGPRs so
// the compiler emits s_load_b32 (SMEM) instead of 32-lane vector loads.
// Gather of h[src] is one coalesced 256B wave read; accumulation uses
// global_atomic_add_f32 (unsafeAtomicAdd, no CAS loop).
// ---------------------------------------------------------------------------
__global__ void __launch_bounds__(256)
gin_scatter(const float* __restrict__ H,
            const int* __restrict__ ei,   // [2, NE]: row 0 = src, row 1 = dst
            float* __restrict__ AGG)
{
    const int gid  = blockIdx.x * 256 + threadIdx.x;
    const int e    = __builtin_amdgcn_readfirstlane(gid >> 5); // wave-uniform
    if (e >= NE) return;
    const int lane = gid & 31;

    const int src = ei[e];        // scalar loads (uniform e)
    const int dst = ei[NE + e];

    const v2f v = *(const v2f*)(H + (size_t)src * FEAT + lane * 2);
    float* d = AGG + (size_t)dst * FEAT + lane * 2;
    unsafeAtomicAdd(d,     v.x);  // -> global_atomic_add_f32
    unsafeAtomicAdd(d + 1, v.y);
}

// ---------------------------------------------------------------------------
// Fused (X + AGG) @ W^T + b (+ReLU) with V_WMMA_F32_16X16X4_F32.
// One wave computes a 16x16 output tile (16 nodes x 16 out-features).
// ISA layouts (CDNA5 ISA, 32-bit):
//   A 16x4 : M = lane%16; lanes 0-15: v0/v1 = K0/K1, lanes 16-31: K2/K3
//   B 4x16 : N = lane%16, same K striping;  B[k][n] = W[n*FEAT + k]
//   C 16x16: VGPR r -> M = r + (lane<16 ? 0 : 8), N = lane%16
// N_OUT == FEAT (hidden layers): all lanes valid -> no masking at all.
// N_OUT  < FEAT (10-class head): tail lanes clamp their W row to row 0 and
// zero the value, so loads stay in-bounds and EXEC is all-1s at every WMMA.
// ---------------------------------------------------------------------------
template <int N_OUT, bool RELU>
__global__ void gin_gemm_wmma(const float* __restrict__ X,
                              const float* __restrict__ AGG,
                              const float* __restrict__ W,   // [N_OUT, FEAT] row-major
                              const float* __restrict__ Bv,  // [N_OUT]
                              float* __restrict__ OUT)       // [NN, N_OUT]
{
    constexpr bool FULL = (N_OUT == FEAT);

    const int lane  = threadIdx.x & 31;
    const int wave  = threadIdx.x >> 5;
    const int half  = lane >> 4;          // 0: lanes 0-15, 1: lanes 16-31
    const int l16   = lane & 15;
    const int node0 = blockIdx.x * 16;

    const int  n      = wave * 16 + l16;
    const bool nvalid = FULL || (n < N_OUT);
    const int  nc     = FULL ? n : (nvalid ? n : 0);
    const float nmask = (FULL || nvalid) ? 1.0f : 0.0f;

    const float* __restrict__ xrow = X   + (size_t)(node0 + l16) * FEAT;
    const float* __restrict__ arow = AGG + (size_t)(node0 + l16) * FEAT;
    const float* __restrict__ wrow = W   + (size_t)nc * FEAT;

    v8f c = {};
#pragma unroll
    for (int k0 = 0; k0 < FEAT; k0 += 4) {
        const int ka = k0 + half * 2;
        v2f a, b;
        a.x = xrow[ka]     + arow[ka];    // fused (x + agg), merges to b64 loads
        a.y = xrow[ka + 1] + arow[ka + 1];
        if constexpr (FULL) {
            b.x = wrow[ka];
            b.y = wrow[ka + 1];
        } else {
            b.x = wrow[ka]     * nmask;
            b.y = wrow[ka + 1] * nmask;
        }
        // 8 args: (neg_a, A, neg_b, B, c_mod, C, reuse_a, reuse_b)
        c = __builtin_amdgcn_wmma_f32_16x16x4_f32(
                false, a, false, b, (short)0, c, false, false);
    }

    if (nvalid) {
        const float bias = Bv[nc];
#pragma unroll
        for (int r = 0; r < 8; ++r) {
            const int row = node0 + half * 8 + r;   // C/D layout
            float v = c[r] + bias;
            if (RELU) v = fmaxf(v, 0.0f);
            OUT[(size_t)row * N_OUT + n] = v;
        }
    }
}

// ---------------------------------------------------------------------------
// Host side: 4 layers, ping-pong hidden buffers in d_ws.
// ws layout: [agg: NN*64 f32][hA: NN*64 f32][hB: NN*64 f32]  (76.8 MB)
// All launches + memsets on `stream` (graph-capture safe).
// ---------------------------------------------------------------------------
extern "C" void kernel_launch(void* const* d_in, const int* in_sizes, int n_in,
                              void* d_out, int out_size, void* d_ws, size_t ws_size,
                              hipStream_t stream) {
    const float* x  = (const float*)d_in[0];
    const int*   ei = (const int*)  d_in[1];
    const float* W1 = (const float*)d_in[2]; const float* b1 = (const float*)d_in[3];
    const float* W2 = (const float*)d_in[4]; const float* b2 = (const float*)d_in[5];
    const float* W3 = (const float*)d_in[6]; const float* b3 = (const float*)d_in[7];
    const float* W4 = (const float*)d_in[8]; const float* b4 = (const float*)d_in[9];
    float* out = (float*)d_out;

    const size_t hbytes = (size_t)NN * FEAT * sizeof(float);
    char* ws  = (char*)d_ws;
    float* agg = (float*)(ws);
    float* hA  = (float*)(ws + hbytes);
    float* hB  = (float*)(ws + 2 * hbytes);

    const dim3 sgrid((NE * 32) / 256);    // 200000 blocks, exact
    const dim3 ggrid(NN / 16);            // 6250 blocks, exact

    // Layer 1: x -> hA
    hipMemsetAsync(agg, 0, hbytes, stream);
    gin_scatter<<<sgrid, 256, 0, stream>>>(x, ei, agg);
    gin_gemm_wmma<FEAT, true><<<ggrid, 128, 0, stream>>>(x, agg, W1, b1, hA);

    // Layer 2: hA -> hB
    hipMemsetAsync(agg, 0, hbytes, stream);
    gin_scatter<<<sgrid, 256, 0, stream>>>(hA, ei, agg);
    gin_gemm_wmma<FEAT, true><<<ggrid, 128, 0, stream>>>(hA, agg, W2, b2, hB);

    // Layer 3: hB -> hA
    hipMemsetAsync(agg, 0, hbytes, stream);
    gin_scatter<<<sgrid, 256, 0, stream>>>(hB, ei, agg);
    gin_gemm_wmma<FEAT, true><<<ggrid, 128, 0, stream>>>(hB, agg, W3, b3, hA);

    // Layer 4: hA -> out, 10 classes, no ReLU, single wave per 16-node tile
    hipMemsetAsync(agg, 0, hbytes, stream);
    gin_scatter<<<sgrid, 256, 0, stream>>>(hA, ei, agg);
    gin_gemm_wmma<CLASSES, false><<<ggrid, 32, 0, stream>>>(hA, agg, W4, b4, out);
}